// LightweightSpatialMamba_86835648790970
// MI455X (gfx1250) — compile-verified
//
#include <hip/hip_runtime.h>
#include <hip/hip_bf16.h>
#include <cstdint>

// ---------------------------------------------------------------------------
// LightweightSpatialMamba for MI455X (gfx1250, wave32).
// GEMMs: V_WMMA_F32_16X16X4_F32 with double-buffered LDS filled by
// GLOBAL_LOAD_ASYNC_TO_LDS_B128 (ASYNCcnt-tracked async copies).
// Scan: VALU, 16 states/thread, Bc staged in LDS.
// ---------------------------------------------------------------------------

typedef __attribute__((ext_vector_type(2))) float v2f;
typedef __attribute__((ext_vector_type(4))) float v4f;
typedef __attribute__((ext_vector_type(8))) float v8f;

#define GEMM_TM 64
#define GEMM_TN 64
#define GEMM_TK 16

__device__ __forceinline__ float siluf(float x) { return x / (1.f + __expf(-x)); }
__device__ __forceinline__ float softplusf(float x) {
  return x > 20.f ? x : __logf(1.f + __expf(x));
}

// Async global->LDS copy of 16 bytes per lane (CDNA5 TDM-class data path).
__device__ __forceinline__ void async_b128_to_lds(void* lds_ptr,
                                                  const float* gptr) {
  // Low 32 bits of a generic shared pointer are the LDS byte offset.
  uint32_t dst = (uint32_t)(uintptr_t)lds_ptr;
  asm volatile("global_load_async_to_lds_b128 %0, %1, off"
               :: "v"(dst), "v"(gptr)
               : "memory");
}
__device__ __forceinline__ void wait_async0() {
  asm volatile("s_wait_asynccnt 0x0" ::: "memory");
}

// ---------------------------------------------------------------------------
// Tiled fp32 GEMM: C[M,N] = A[M,K] @ B[K,N]  (row-major, arbitrary ld).
// epilogue 0: store; epilogue 1: softplus(acc + bias[col]).
// Requires M % 64 == 0, K % 16 == 0, N % 4 == 0 (cols beyond N guarded).
// ---------------------------------------------------------------------------
__global__ __launch_bounds__(128)
void gemm_wmma_f32(const float* __restrict__ A, const float* __restrict__ B,
                   float* __restrict__ C, int M, int N, int K,
                   int lda, int ldb, int ldc, int epilogue,
                   const float* __restrict__ bias) {
  __shared__ __align__(16) float As[2][GEMM_TM * GEMM_TK];  // 2 x 64x16
  __shared__ __align__(16) float Bs[2][GEMM_TK * GEMM_TN];  // 2 x 16x64

  const int tid  = threadIdx.x;      // 0..127
  const int wave = tid >> 5;         // 0..3
  const int lane = tid & 31;
  const int lane16   = lane & 15;
  const int laneHalf = lane >> 4;    // 0: lanes 0-15, 1: lanes 16-31

  const int rowBase = blockIdx.y * GEMM_TM;
  const int colBase = blockIdx.x * GEMM_TN;

  v8f acc[4];
#pragma unroll
  for (int t = 0; t < 4; ++t)
#pragma unroll
    for (int i = 0; i < 8; ++i) acc[t][i] = 0.f;

  // Issue async copies for one 64x16 A tile + 16x64 B tile (4 ops/thread).
  auto issueTile = [&](int buf, int k0) {
#pragma unroll
    for (int i = tid; i < (GEMM_TM * GEMM_TK) / 4; i += 128) {
      int r  = i >> 2;                 // 4 float4 per 16-float A row
      int c4 = (i & 3) * 4;
      async_b128_to_lds(&As[buf][i * 4],
                        A + (size_t)(rowBase + r) * lda + k0 + c4);
    }
#pragma unroll
    for (int i = tid; i < (GEMM_TK * GEMM_TN) / 4; i += 128) {
      int r  = i >> 4;                 // 16 float4 per 64-float B row
      int c4 = (i & 15) * 4;
      async_b128_to_lds(&Bs[buf][i * 4],
                        B + (size_t)(k0 + r) * ldb + colBase + c4);
    }
  };

  // 16 WMMAs on one staged K-chunk.
  auto computeChunk = [&](int buf) {
    const float* as = As[buf];
    const float* bs = Bs[buf];
#pragma unroll
    for (int kk = 0; kk < GEMM_TK; kk += 4) {
      // A frag 16x4: lanes 0-15 hold K=kk..kk+1, lanes 16-31 K=kk+2..kk+3
      v2f afrag;
      afrag[0] = as[(wave * 16 + lane16) * GEMM_TK + kk + laneHalf * 2 + 0];
      afrag[1] = as[(wave * 16 + lane16) * GEMM_TK + kk + laneHalf * 2 + 1];
#pragma unroll
      for (int nt = 0; nt < 4; ++nt) {
        // B frag 4x16: VGPR0 rows K=kk (lanes 0-15) / K=kk+2 (lanes 16-31)
        v2f bfrag;
        bfrag[0] = bs[(kk + laneHalf * 2 + 0) * GEMM_TN + nt * 16 + lane16];
        bfrag[1] = bs[(kk + laneHalf * 2 + 1) * GEMM_TN + nt * 16 + lane16];
        acc[nt] = __builtin_amdgcn_wmma_f32_16x16x4_f32(
            false, afrag, false, bfrag, (short)0, acc[nt], false, false);
      }
    }
  };

  const int  nk       = K / GEMM_TK;
  const bool fullTile = (colBase + GEMM_TN) <= N;

  if (fullTile) {
    // Async double-buffered pipeline: copy engine fills buf k+1 while
    // WMMAs consume buf k. One barrier per chunk.
    issueTile(0, 0);
    for (int k = 0; k < nk; ++k) {
      const int cur = k & 1;
      wait_async0();        // own chunk-k copies landed in LDS
      __syncthreads();      // all waves' chunk-k data visible; prev buf free
      if (k + 1 < nk) issueTile(cur ^ 1, (k + 1) * GEMM_TK);
      computeChunk(cur);
    }
  } else {
    // Guarded synchronous staging (skinny-N tiles only).
    for (int k0 = 0; k0 < K; k0 += GEMM_TK) {
      __syncthreads();
#pragma unroll
      for (int i = tid; i < (GEMM_TM * GEMM_TK) / 4; i += 128) {
        int r  = i >> 2;
        int c4 = (i & 3) * 4;
        ((v4f*)As[0])[i] =
            *(const v4f*)(A + (size_t)(rowBase + r) * lda + k0 + c4);
      }
#pragma unroll
      for (int i = tid; i < (GEMM_TK * GEMM_TN) / 4; i += 128) {
        int r  = i >> 4;
        int c4 = (i & 15) * 4;
        int gcol = colBase + c4;
        v4f v;
        if (gcol < N) {
          v = *(const v4f*)(B + (size_t)(k0 + r) * ldb + gcol);
        } else {
          v[0] = v[1] = v[2] = v[3] = 0.f;
        }
        ((v4f*)Bs[0])[i] = v;
      }
      __syncthreads();
      computeChunk(0);
    }
  }

  // Store: VGPR r -> (M = r + laneHalf*8, N = lane16) of each 16x16 tile.
#pragma unroll
  for (int nt = 0; nt < 4; ++nt) {
    int col = colBase + nt * 16 + lane16;
    if (col >= N) continue;
#pragma unroll
    for (int r = 0; r < 8; ++r) {
      int row = rowBase + wave * 16 + r + laneHalf * 8;
      float v = acc[nt][r];
      if (epilogue == 1) v = softplusf(v + bias[col]);
      C[(size_t)row * ldc + col] = v;
    }
  }
}

// ---------------------------------------------------------------------------
// Permutation tables for the 4 scan directions (reads H/W from device mem).
// ---------------------------------------------------------------------------
__global__ void init_perms_kernel(int* __restrict__ perms,
                                  const int* __restrict__ hptr,
                                  const int* __restrict__ wptr) {
  const int H = *hptr, W = *wptr, N = H * W;
  int l = threadIdx.x;
  if (l < N) {
    perms[0 * N + l] = l;                       // forward
    perms[1 * N + l] = N - 1 - l;               // flip(H,W)
    int i = l % H, j = l / H;                   // transpose: l = j*H + i
    perms[2 * N + l] = i * W + j;
  }
  if (threadIdx.x == 0) {                       // diagonal order
    int idx = 0;
    for (int off = -(H - 1); off < W; ++off)
      for (int i = 0; i < H; ++i) {
        int j = i + off;
        if (0 <= j && j < W) perms[3 * N + (idx++)] = i * W + j;
      }
  }
}

__global__ void zero_kernel(float* __restrict__ p, size_t n) {
  size_t i = (size_t)blockIdx.x * blockDim.x + threadIdx.x;
  if (i < n) p[i] = 0.f;
}

// XP[b, l, :] = X[b, perm[l], :]
__global__ void gather_rows_kernel(float* __restrict__ XP,
                                   const float* __restrict__ X,
                                   const int* __restrict__ perm,
                                   int L, int E, size_t total) {
  size_t i = (size_t)blockIdx.x * blockDim.x + threadIdx.x;
  if (i >= total) return;
  int e = (int)(i % E);
  size_t row = i / E;
  int b = (int)(row / L), l = (int)(row % L);
  XP[i] = X[((size_t)b * L + perm[l]) * E + e];
}

// Depthwise causal conv (k=4) over L within each batch + SiLU.
// xz has leading-dim 2E (xi = first E columns).
__global__ void conv_silu_kernel(const float* __restrict__ xz,
                                 const float* __restrict__ conv_w,
                                 float* __restrict__ xc,
                                 int L, int E, size_t total) {
  size_t i = (size_t)blockIdx.x * blockDim.x + threadIdx.x;
  if (i >= total) return;
  int e = (int)(i % E);
  size_t row = i / E;
  int b = (int)(row / L), l = (int)(row % L);
  float acc = 0.f;
#pragma unroll
  for (int t = 0; t < 4; ++t) {
    int ll = l - 3 + t;
    if (ll >= 0)
      acc += conv_w[e * 4 + t] * xz[((size_t)b * L + ll) * (2 * E) + e];
  }
  xc[i] = siluf(acc);
}

// ---------------------------------------------------------------------------
// Selective scan: one block per batch, thread = channel, 16 states in regs.
// Accumulates dir_w[d] * y * silu(z) scattered through perm into U.
// ---------------------------------------------------------------------------
__global__ __launch_bounds__(1024)
void scan_kernel(const float* __restrict__ dt, const float* __restrict__ dtbc,
                 const float* __restrict__ xc, const float* __restrict__ xz,
                 const float* __restrict__ A_log, const int* __restrict__ perm,
                 const float* __restrict__ dir_w, int dirIdx,
                 float* __restrict__ U, int L, int E) {
  const int b = blockIdx.x;
  const int c = threadIdx.x;  // 0..E-1 (E == blockDim.x == 1024)
  __shared__ float BcS[64 * 16];
  __shared__ int permS[64];

  const float w = dir_w[dirIdx];
  float An[16], h[16];
#pragma unroll
  for (int s = 0; s < 16; ++s) {
    An[s] = -__expf(A_log[c * 16 + s]);
    h[s]  = 0.f;
  }

  for (int l0 = 0; l0 < L; l0 += 64) {
    __syncthreads();
    {  // stage Bc[l0..l0+63][0..15] (dtbc cols 16..31); 1024 threads, 1 each
      int r = c >> 4, s = c & 15;
      BcS[c] = dtbc[((size_t)b * L + l0 + r) * 32 + 16 + s];
      if (c < 64) permS[c] = perm[l0 + c];
    }
    __syncthreads();

    for (int li = 0; li < 64; ++li) {
      size_t row = (size_t)b * L + l0 + li;
      float dtv = dt[row * E + c];
      float xv  = xc[row * E + c];
      float zv  = xz[row * (size_t)(2 * E) + E + c];
      float y = 0.f;
#pragma unroll
      for (int s = 0; s < 16; ++s) {
        h[s] = h[s] * __expf(An[s] * dtv) + BcS[li * 16 + s] * xv;
        y += h[s];
      }
      size_t orow = ((size_t)b * L + permS[li]) * E + c;
      U[orow] += w * y * siluf(zv);
    }
  }
}

// Row-wise LayerNorm in place. One block (256 thr) per row.
__global__ __launch_bounds__(256)
void layernorm_kernel(float* __restrict__ x, const float* __restrict__ g,
                      const float* __restrict__ bta, int E) {
  __shared__ float r1[256], r2[256];
  const size_t row = blockIdx.x;
  const int tid = threadIdx.x;
  float s1 = 0.f, s2 = 0.f;
  for (int e = tid; e < E; e += 256) {
    float v = x[row * E + e];
    s1 += v; s2 += v * v;
  }
  r1[tid] = s1; r2[tid] = s2;
  __syncthreads();
  for (int off = 128; off > 0; off >>= 1) {
    if (tid < off) { r1[tid] += r1[tid + off]; r2[tid] += r2[tid + off]; }
    __syncthreads();
  }
  const float mu  = r1[0] / E;
  const float var = r2[0] / E - mu * mu;
  const float inv = rsqrtf(var + 1e-5f);
  for (int e = tid; e < E; e += 256)
    x[row * E + e] = (x[row * E + e] - mu) * inv * g[e] + bta[e];
}

// mean over sequence: meanb[b,e] = (1/L) sum_l vf[b,l,e]
__global__ void mean_kernel(float* __restrict__ meanb,
                            const float* __restrict__ vf, int B, int L, int E) {
  int i = blockIdx.x * blockDim.x + threadIdx.x;
  if (i >= B * E) return;
  int b = i / E, e = i % E;
  float s = 0.f;
  for (int l = 0; l < L; ++l) s += vf[((size_t)b * L + l) * E + e];
  meanb[i] = s / (float)L;
}

// gate[b] = sigmoid( silu(mean @ Wg1) @ Wg2 )
__global__ __launch_bounds__(256)
void gate_kernel(const float* __restrict__ meanb, const float* __restrict__ Wg1,
                 const float* __restrict__ Wg2, float* __restrict__ gate,
                 int E, int Eq) {
  __shared__ float red[256];
  const int b = blockIdx.x, j = threadIdx.x;
  float acc = 0.f;
  for (int e = 0; e < E; ++e) acc += meanb[b * E + e] * Wg1[e * Eq + j];
  red[j] = siluf(acc) * Wg2[j];
  __syncthreads();
  for (int off = 128; off > 0; off >>= 1) {
    if (j < off) red[j] += red[j + off];
    __syncthreads();
  }
  if (j == 0) gate[b] = 1.f / (1.f + __expf(-red[0]));
}

__global__ void final_kernel(float* __restrict__ out,
                             const float* __restrict__ vf,
                             const float* __restrict__ enh,
                             const float* __restrict__ gate,
                             size_t perBatch, size_t total) {
  size_t i = (size_t)blockIdx.x * blockDim.x + threadIdx.x;
  if (i >= total) return;
  out[i] = vf[i] + gate[i / perBatch] * enh[i];
}

// ---------------------------------------------------------------------------
static inline void launch_gemm(const float* A, const float* B, float* C,
                               int M, int N, int K, int lda, int ldb, int ldc,
                               int ep, const float* bias, hipStream_t stream) {
  dim3 grid((N + GEMM_TN - 1) / GEMM_TN, M / GEMM_TM);
  gemm_wmma_f32<<<grid, 128, 0, stream>>>(A, B, C, M, N, K, lda, ldb, ldc, ep, bias);
}

extern "C" void kernel_launch(void* const* d_in, const int* in_sizes, int n_in,
                              void* d_out, int out_size, void* d_ws, size_t ws_size,
                              hipStream_t stream) {
  const int E = 1024, L = 1024, EQ = 256;
  const float* vf      = (const float*)d_in[0];
  const float* W_input = (const float*)d_in[1];
  const float* W_in    = (const float*)d_in[2];
  const float* conv_w  = (const float*)d_in[3];
  const float* W_xproj = (const float*)d_in[4];
  const float* W_dt    = (const float*)d_in[5];
  const float* b_dt    = (const float*)d_in[6];
  const float* A_log   = (const float*)d_in[7];
  const float* W_outm  = (const float*)d_in[8];
  const float* dir_w   = (const float*)d_in[9];
  const float* ln_g    = (const float*)d_in[10];
  const float* ln_b    = (const float*)d_in[11];
  const float* W_outp  = (const float*)d_in[12];
  const float* Wg1     = (const float*)d_in[13];
  const float* Wg2     = (const float*)d_in[14];
  const int*   hptr    = (const int*)d_in[15];
  const int*   wptr    = (const int*)d_in[16];

  const int B = in_sizes[0] / (L * E);           // 8
  const size_t Mr = (size_t)B * L;               // 8192 rows
  const int Mi = (int)Mr;

  float* ws = (float*)d_ws;
  size_t off = 0;
  float* X     = ws + off; off += Mr * E;        // vf @ W_input
  float* XC    = ws + off; off += Mr * E;        // gathered xp, then conv out
  float* XZ    = ws + off; off += Mr * 2 * E;    // xp @ W_in
  float* DTBC  = ws + off; off += Mr * 32;       // xc @ W_xproj
  float* DT    = ws + off; off += Mr * E;        // softplus(dtbc@W_dt + b)
  float* U     = ws + off; off += Mr * E;        // sum_d dirw * scatter(y*silu(z))
  float* FUSED = ws + off; off += Mr * E;
  float* ENH   = ws + off; off += Mr * E;
  float* MEANB = ws + off; off += (size_t)B * E;
  float* GATE  = ws + off; off += (size_t)B;
  int*   PERMS = (int*)(ws + off);               // 4 * L ints

  const size_t totBE = Mr * E;
  const int blk = 256;
  const int gridBE = (int)((totBE + blk - 1) / blk);

  init_perms_kernel<<<1, 1024, 0, stream>>>(PERMS, hptr, wptr);

  // X = vf @ W_input
  launch_gemm(vf, W_input, X, Mi, E, E, E, E, E, 0, nullptr, stream);

  zero_kernel<<<gridBE, blk, 0, stream>>>(U, totBE);

  for (int d = 0; d < 4; ++d) {
    const int* perm = PERMS + d * L;
    gather_rows_kernel<<<gridBE, blk, 0, stream>>>(XC, X, perm, L, E, totBE);
    // XZ = XP @ W_in   (2048 wide)
    launch_gemm(XC, W_in, XZ, Mi, 2 * E, E, E, 2 * E, 2 * E, 0, nullptr, stream);
    // XC = silu(conv1d(xi))   (reuses XP buffer)
    conv_silu_kernel<<<gridBE, blk, 0, stream>>>(XZ, conv_w, XC, L, E, totBE);
    // DTBC = XC @ W_xproj   (N = 32, guarded sync path)
    launch_gemm(XC, W_xproj, DTBC, Mi, 32, E, E, 32, 32, 0, nullptr, stream);
    // DT = softplus(DTBC[:, :16] @ W_dt + b_dt)   (K = 16, lda = 32)
    launch_gemm(DTBC, W_dt, DT, Mi, E, 16, 32, E, E, 1, b_dt, stream);
    // scan + gate-by-silu(z) + weighted scatter into U
    scan_kernel<<<B, 1024, 0, stream>>>(DT, DTBC, XC, XZ, A_log, perm,
                                        dir_w, d, U, L, E);
  }

  // FUSED = U @ W_outm  (shared across directions by linearity)
  launch_gemm(U, W_outm, FUSED, Mi, E, E, E, E, E, 0, nullptr, stream);
  layernorm_kernel<<<Mi, 256, 0, stream>>>(FUSED, ln_g, ln_b, E);
  // ENH = FUSED @ W_outproj
  launch_gemm(FUSED, W_outp, ENH, Mi, E, E, E, E, E, 0, nullptr, stream);

  mean_kernel<<<(B * E + blk - 1) / blk, blk, 0, stream>>>(MEANB, vf, B, L, E);
  gate_kernel<<<B, 256, 0, stream>>>(MEANB, Wg1, Wg2, GATE, E, EQ);
  final_kernel<<<gridBE, blk, 0, stream>>>((float*)d_out, vf, ENH, GATE,
                                           (size_t)L * E, totBE);
}